// SimpleGNN_24103356465666
// MI455X (gfx1250) — compile-verified
//
#include <hip/hip_runtime.h>
#include <hip/hip_bf16.h>

// ---------------------------------------------------------------------------
// SimpleGNN (2x GCNConv + mean-pool + linear head) for MI455X / gfx1250.
// Dense transforms use V_WMMA_F32_16X16X4_F32 (exact fp32, wave32 tiles).
// B (weight) fragments are register-resident and amortized over 5 row tiles
// per block; A panel staged through LDS. Sparse aggregation uses global f32
// atomics; self-loops folded into the elementwise epilogue (no atomics).
// ---------------------------------------------------------------------------

typedef __attribute__((ext_vector_type(2))) float v2f;
typedef __attribute__((ext_vector_type(8))) float v8f;

#define D_FEAT 128
#define ROW_TILES 5                      // 16-row tiles per block
#define ROWS_PER_BLK (ROW_TILES * 16)    // 80 rows; 100000 % 80 == 0

// ---- degree / normalization -----------------------------------------------
__global__ __launch_bounds__(256) void k_init_deg(float* deg, int N) {
    int i = blockIdx.x * 256 + threadIdx.x;
    if (i < N) deg[i] = 1.0f;               // self-loop contributes 1
}

__global__ __launch_bounds__(256) void k_edge_deg(const long long* __restrict__ dst,
                                                  float* deg, int E) {
    int e = blockIdx.x * 256 + threadIdx.x;
    if (e < E) atomicAdd(&deg[(int)dst[e]], 1.0f);
}

__global__ __launch_bounds__(256) void k_rsqrt(float* d, int N) {
    int i = blockIdx.x * 256 + threadIdx.x;
    if (i < N) d[i] = rsqrtf(d[i]);         // deg >= 1 always (self-loop)
}

// ---- generic zero ----------------------------------------------------------
__global__ __launch_bounds__(256) void k_zero(float* p, size_t n) {
    size_t i = (size_t)blockIdx.x * 256 + threadIdx.x;
    if (i < n) p[i] = 0.0f;
}

// ---- dense GEMM: T[i,:] = dis[i] * (X[i,:] @ W)  via f32 WMMA --------------
// Block = 256 threads = 8 waves. Block owns 80 node-rows x 128 out-cols.
// Each wave owns one 16-wide column group; its 32 B fragments (v2f each) are
// loaded ONCE into registers and reused across 5 row tiles (160 WMMAs/wave).
__global__ __launch_bounds__(256) void k_gemm_wmma(const float* __restrict__ X,
                                                   const float* __restrict__ W,
                                                   const float* __restrict__ dis,
                                                   float* __restrict__ T, int N) {
    __shared__ float lx[ROWS_PER_BLK * D_FEAT];   // 40 KB A panel

    const int nodeBase = blockIdx.x * ROWS_PER_BLK;
    const int tid  = threadIdx.x;

    // Cooperative load of the 80x128 panel: 2560 float4 -> 10 per thread.
    const float4* xg = (const float4*)(X + (size_t)nodeBase * D_FEAT);
    float4* ls = (float4*)lx;
    #pragma unroll
    for (int j = 0; j < (ROWS_PER_BLK * D_FEAT / 4) / 256; ++j)
        ls[tid + j * 256] = xg[tid + j * 256];

    const int lane = tid & 31;
    const int wave = tid >> 5;
    const int l    = lane & 15;             // N-col within tile / M-row for A
    const int hi   = lane >> 4;             // lane half selects K-pair / M+8
    const int col  = wave * 16 + l;         // global output column

    // Preload B fragments for this wave's column group (register-resident).
    v2f bfrag[32];
    #pragma unroll
    for (int kk = 0; kk < 32; ++kk) {
        const int k0 = kk * 4 + hi * 2;
        bfrag[kk].x = W[(size_t)k0 * D_FEAT + col];
        bfrag[kk].y = W[(size_t)(k0 + 1) * D_FEAT + col];
    }
    __syncthreads();

    #pragma unroll
    for (int t = 0; t < ROW_TILES; ++t) {
        const float* lt = lx + t * 16 * D_FEAT;
        v8f c = {};
        #pragma unroll
        for (int kk = 0; kk < 32; ++kk) {
            const int k0 = kk * 4 + hi * 2;
            v2f a;
            // A fragment (ISA 32-bit A 16x4): lane row l, K = k0, k0+1
            a.x = lt[l * D_FEAT + k0];
            a.y = lt[l * D_FEAT + k0 + 1];
            c = __builtin_amdgcn_wmma_f32_16x16x4_f32(
                    /*neg_a=*/false, a, /*neg_b=*/false, bfrag[kk],
                    /*c_mod=*/(short)0, c, /*reuse_a=*/false, /*reuse_b=*/false);
        }
        // Epilogue: scale each output row by dis[row] (src-side GCN norm).
        #pragma unroll
        for (int v = 0; v < 8; ++v) {
            const int row = nodeBase + t * 16 + v + hi * 8;  // C/D: M=v / M=v+8
            T[(size_t)row * D_FEAT + col] = c[v] * dis[row];
        }
    }
}

// ---- edge scatter-add: U[dst,:] += T[src,:]  (one wave per edge) -----------
__global__ __launch_bounds__(256) void k_edge_agg(const float* __restrict__ T,
                                                  const long long* __restrict__ src,
                                                  const long long* __restrict__ dst,
                                                  float* __restrict__ U, int E) {
    const int e = (int)(((size_t)blockIdx.x * 256 + threadIdx.x) >> 5);
    if (e >= E) return;
    const int lane = threadIdx.x & 31;
    const int s = (int)src[e];
    const int d = (int)dst[e];
    const float4 m = ((const float4*)(T + (size_t)s * D_FEAT))[lane];
    float* ud = U + (size_t)d * D_FEAT + lane * 4;
    atomicAdd(ud + 0, m.x);
    atomicAdd(ud + 1, m.y);
    atomicAdd(ud + 2, m.z);
    atomicAdd(ud + 3, m.w);
}

// ---- epilogue: A = relu(dis[d]*(U + T_self) + b) ---------------------------
__global__ __launch_bounds__(256) void k_post(const float* __restrict__ U,
                                              const float* __restrict__ T,
                                              const float* __restrict__ dis,
                                              const float* __restrict__ b,
                                              float* __restrict__ A, int N) {
    const size_t idx = (size_t)blockIdx.x * 256 + threadIdx.x;
    if (idx >= (size_t)N * D_FEAT) return;
    const int f = (int)(idx & (D_FEAT - 1));
    const int n = (int)(idx >> 7);
    const float h = dis[n] * (U[idx] + T[idx]) + b[f];
    A[idx] = fmaxf(h, 0.0f);
}

// ---- global mean pool (sums + counts) --------------------------------------
__global__ __launch_bounds__(256) void k_pool(const float* __restrict__ A,
                                              const long long* __restrict__ batch,
                                              float* __restrict__ sums,
                                              float* __restrict__ cnts, int N) {
    const int n = (int)(((size_t)blockIdx.x * 256 + threadIdx.x) >> 5);
    if (n >= N) return;
    const int lane = threadIdx.x & 31;
    const int g = (int)batch[n];
    const float4 v = ((const float4*)(A + (size_t)n * D_FEAT))[lane];
    float* sg = sums + (size_t)g * D_FEAT + lane * 4;
    atomicAdd(sg + 0, v.x);
    atomicAdd(sg + 1, v.y);
    atomicAdd(sg + 2, v.z);
    atomicAdd(sg + 3, v.w);
    if (lane == 0) atomicAdd(&cnts[g], 1.0f);
}

// ---- linear head: out[g,c] = bl[c] + mean_g . Wl[:,c] ----------------------
__global__ __launch_bounds__(256) void k_head(const float* __restrict__ sums,
                                              const float* __restrict__ cnts,
                                              const float* __restrict__ Wl,
                                              const float* __restrict__ bl,
                                              float* __restrict__ out,
                                              int G, int C) {
    const int idx = blockIdx.x * 256 + threadIdx.x;
    if (idx >= G * C) return;
    const int g = idx / C;
    const int c = idx % C;
    const float inv = 1.0f / fmaxf(cnts[g], 1.0f);
    float acc = bl[c];
    for (int f = 0; f < D_FEAT; ++f)
        acc += sums[(size_t)g * D_FEAT + f] * inv * Wl[f * C + c];
    out[idx] = acc;
}

extern "C" void kernel_launch(void* const* d_in, const int* in_sizes, int n_in,
                              void* d_out, int out_size, void* d_ws, size_t ws_size,
                              hipStream_t stream) {
    const float*     x     = (const float*)d_in[0];
    const long long* ei    = (const long long*)d_in[1];   // [2,E] int64
    const long long* batch = (const long long*)d_in[2];   // [N]   int64
    const float*     W1    = (const float*)d_in[3];
    const float*     b1    = (const float*)d_in[4];
    const float*     W2    = (const float*)d_in[5];
    const float*     b2    = (const float*)d_in[6];
    const float*     Wl    = (const float*)d_in[7];
    const float*     bl    = (const float*)d_in[8];
    float*           out   = (float*)d_out;

    const int N = in_sizes[0] / D_FEAT;     // 100000
    const int E = in_sizes[1] / 2;          // 600000
    const int C = 2;
    const int G = out_size / C;             // 512

    const long long* src = ei;              // edge_index[0]
    const long long* dst = ei + E;          // edge_index[1]

    // Workspace layout (floats):
    float* bufA = (float*)d_ws;                       // [N,128] gemm out (T)
    float* bufB = bufA + (size_t)N * D_FEAT;          // [N,128] agg / activ.
    float* dis  = bufB + (size_t)N * D_FEAT;          // [N]   deg -> rsqrt
    float* sums = dis + N;                            // [G,128]
    float* cnts = sums + (size_t)G * D_FEAT;          // [G]

    const int nodeBlk  = (N + 255) / 256;
    const int edgeBlk  = (E + 255) / 256;
    const int gemmBlk  = N / ROWS_PER_BLK;                        // 1250
    const int warpEdge = (int)(((size_t)E * 32 + 255) / 256);     // 75000
    const int elemBlk  = (int)(((size_t)N * D_FEAT + 255) / 256); // 50000
    const int warpNode = (int)(((size_t)N * 32 + 255) / 256);     // 12500

    // --- GCN normalization: dis = rsqrt(1 + indeg) ---
    k_init_deg<<<nodeBlk, 256, 0, stream>>>(dis, N);
    k_edge_deg<<<edgeBlk, 256, 0, stream>>>(dst, dis, E);
    k_rsqrt<<<nodeBlk, 256, 0, stream>>>(dis, N);

    // --- Layer 1 ---
    k_gemm_wmma<<<gemmBlk, 256, 0, stream>>>(x, W1, dis, bufA, N);
    k_zero<<<elemBlk, 256, 0, stream>>>(bufB, (size_t)N * D_FEAT);
    k_edge_agg<<<warpEdge, 256, 0, stream>>>(bufA, src, dst, bufB, E);
    k_post<<<elemBlk, 256, 0, stream>>>(bufB, bufA, dis, b1, bufB, N);

    // --- Layer 2 ---
    k_gemm_wmma<<<gemmBlk, 256, 0, stream>>>(bufB, W2, dis, bufA, N);
    k_zero<<<elemBlk, 256, 0, stream>>>(bufB, (size_t)N * D_FEAT);
    k_edge_agg<<<warpEdge, 256, 0, stream>>>(bufA, src, dst, bufB, E);
    k_post<<<elemBlk, 256, 0, stream>>>(bufB, bufA, dis, b2, bufB, N);

    // --- Mean pool + head ---
    k_zero<<<(G * D_FEAT + 255) / 256, 256, 0, stream>>>(sums, (size_t)G * D_FEAT);
    k_zero<<<(G + 255) / 256, 256, 0, stream>>>(cnts, (size_t)G);
    k_pool<<<warpNode, 256, 0, stream>>>(bufB, batch, sums, cnts, N);
    k_head<<<(G * C + 255) / 256, 256, 0, stream>>>(sums, cnts, Wl, bl, out, G, C);
}